// ShapeletNet_82592221102596
// MI455X (gfx1250) — compile-verified
//
#include <hip/hip_runtime.h>

// CDNA5 / gfx1250: wave32, WMMA f32 16x16x4, async global->LDS copies.
typedef __attribute__((ext_vector_type(2))) float v2f;
typedef __attribute__((ext_vector_type(4))) float v4f;
typedef __attribute__((ext_vector_type(8))) float v8f;
typedef __attribute__((ext_vector_type(4))) int   v4i;

// pointer-to-AS1-int4 / pointer-to-AS3-int4, matching the builtin's prototype
typedef __attribute__((address_space(1))) v4i* gbl_v4i_p;
typedef __attribute__((address_space(3))) v4i* lds_v4i_p;

#define DCH   64                 // feature dim D
#define MQ    32                 // shapelet length M
#define PW    256                // window starts per chunk (== blockDim)
#define TLEN  (PW + MQ)          // 288 t-columns staged per chunk
#define TPAD  296                // LDS row stride (floats); 2*296 % 64 == 16 -> conflict-free B reads
#define BLOCK 256                // 8 wave32 waves
#define BIGF  1e30f

// --- CDNA5 async global->LDS copy (GLOBAL_LOAD_ASYNC_TO_LDS_B128, ASYNCcnt),
//     with a portable fallback if this toolchain lacks the builtins.
static __device__ __forceinline__ void async_copy16(const float* __restrict__ g,
                                                    float* __restrict__ l) {
#if __has_builtin(__builtin_amdgcn_global_load_async_to_lds_b128)
  __builtin_amdgcn_global_load_async_to_lds_b128(
      (gbl_v4i_p)(void*)(const_cast<float*>(g)),
      (lds_v4i_p)(void*)l, /*offset=*/0, /*cpol=*/0);
#else
  *(v4f*)l = *(const v4f*)g;
#endif
}

static __device__ __forceinline__ void async_wait0() {
#if __has_builtin(__builtin_amdgcn_s_wait_asynccnt)
  __builtin_amdgcn_s_wait_asynccnt(0);
#else
  asm volatile("s_wait_asynccnt 0" ::: "memory");
#endif
}

// Phase 1: per-segment min of sliding squared distance (minus the constant ||q||^2,
// which cancels exactly in the later min-max scaling).
// dist[p] = sum_j s[p+j] - 2 * sum_j C[j, p+j],  C = Q^T (32x64) @ X (64xT) via WMMA.
__global__ __launch_bounds__(BLOCK) void shapelet_seg_min(
    const float* __restrict__ X, const float* __restrict__ Q,
    const int* __restrict__ cumlens, float* __restrict__ mins, int T)
{
  __shared__ __align__(16) float Xc[DCH * TPAD];   // X tile  [64][TLEN]   (~76 KB)
  __shared__ __align__(16) float Cs[MQ * TPAD];    // C tile  [32][TLEN]   (~38 KB)
  __shared__ float sarr[TLEN];                     // column sums of squares
  __shared__ float red[BLOCK];

  const int tid  = threadIdx.x;
  const int lane = tid & 31;
  const int wave = tid >> 5;
  const int seg  = blockIdx.x;

  const int c0   = cumlens[seg];
  const int c1   = cumlens[seg + 1];
  const int L    = c1 - c0;
  const int numW = L - MQ + 1;       // may be <= 0 -> segment min stays BIG

  // ---- per-wave A fragment: Q^T row-tile r, constant for the whole kernel.
  // ISA A 16x4 layout: lane&15 = M row, lane>>4 selects K-pair, vgpr = K&1.
  const int half2 = (lane >> 4) << 1;
  const int r     = wave & 1;                       // j-tile: rows [16r, 16r+16)
  const int jA    = r * 16 + (lane & 15);
  v2f a[16];
#pragma unroll
  for (int kk = 0; kk < 16; ++kk) {
    const int d0 = kk * 4 + half2;
    a[kk].x = Q[(d0 + 0) * MQ + jA];                // Q^T[j, d] = query[d, j]
    a[kk].y = Q[(d0 + 1) * MQ + jA];
  }

  float myMin = BIGF;

  for (int ps = 0; ps < numW; ps += PW) {
    const int pwc = min(PW,  numW - ps);            // valid window starts this chunk
    const int lim = min(TLEN, L - ps);              // valid t-columns this chunk

    __syncthreads();                                // previous chunk fully consumed

    // ---- stage A: global -> LDS
    const float* xbase = X + (size_t)(c0 + ps);
    if (lim == TLEN && (((c0 + ps) & 3) == 0) && ((T & 3) == 0)) {
      // interior chunk: direct-to-LDS async copies, no VGPR staging
      for (int i = tid; i < DCH * (TLEN / 4); i += BLOCK) {
        const int d = i / (TLEN / 4), tq = i % (TLEN / 4);
        async_copy16(xbase + (size_t)d * T + (size_t)tq * 4, &Xc[d * TPAD + tq * 4]);
      }
      async_wait0();                                // our issues complete before barrier
    } else {
      // boundary chunk: guarded scalar copy with zero padding
      for (int i = tid; i < DCH * TLEN; i += BLOCK) {
        const int d = i / TLEN, tt = i % TLEN;
        Xc[d * TPAD + tt] = (tt < lim) ? xbase[(size_t)d * T + tt] : 0.0f;
      }
    }
    __syncthreads();

    // ---- stage B: s[tt] = sum_d X[d,tt]^2
    for (int tt = tid; tt < TLEN; tt += BLOCK) {
      float s = 0.f;
#pragma unroll 8
      for (int d = 0; d < DCH; ++d) { const float x = Xc[d * TPAD + tt]; s += x * x; }
      sarr[tt] = s;
    }

    // ---- stage C: C = Q^T @ Xc via V_WMMA_F32_16X16X4_F32 (16 K-steps of 4)
    const int nColTiles = TLEN / 16;                // 18
    for (int c = (wave >> 1); c < nColTiles; c += 4) {
      const int coln = c * 16 + (lane & 15);
      v8f acc = {};
#pragma unroll
      for (int kk = 0; kk < 16; ++kk) {
        const int d0 = kk * 4 + half2;
        v2f b;                                      // B 4x16: mirrors A with M<->N
        b.x = Xc[(d0 + 0) * TPAD + coln];
        b.y = Xc[(d0 + 1) * TPAD + coln];
        acc = __builtin_amdgcn_wmma_f32_16x16x4_f32(
            /*neg_a=*/false, a[kk], /*neg_b=*/false, b,
            /*c_mod=*/(short)0, acc, /*reuse_a=*/false, /*reuse_b=*/false);
      }
      // D 16x16 layout: VGPR g -> row g (lanes 0-15) / row g+8 (lanes 16-31)
      const int jbase = r * 16 + ((lane >> 4) << 3);
#pragma unroll
      for (int g = 0; g < 8; ++g)
        Cs[(jbase + g) * TPAD + coln] = acc[g];
    }
    __syncthreads();

    // ---- stage D: diagonal gather, one window per thread (conflict-free: addr = tid + j*(TPAD+1))
    if (tid < pwc) {
      float xx = 0.f, qx = 0.f;
#pragma unroll
      for (int j = 0; j < MQ; ++j) {
        xx += sarr[tid + j];
        qx += Cs[j * TPAD + tid + j];
      }
      myMin = fminf(myMin, xx - 2.0f * qx);
    }
  }

  // ---- block min reduction
  red[tid] = myMin;
  __syncthreads();
  for (int s = BLOCK / 2; s > 0; s >>= 1) {
    if (tid < s) red[tid] = fminf(red[tid], red[tid + s]);
    __syncthreads();
  }
  if (tid == 0) mins[seg] = red[0];
}

// Phase 2: global min/max over segment minima, min-max scale, Linear(1,2).
__global__ __launch_bounds__(1024) void shapelet_finalize(
    const float* __restrict__ mins, const float* __restrict__ lin_w,
    const float* __restrict__ lin_b, float* __restrict__ out, int nseg)
{
  __shared__ float slo[1024], shi[1024];
  const int tid = threadIdx.x;
  float lo = BIGF, hi = -BIGF;
  for (int i = tid; i < nseg; i += 1024) {
    const float v = mins[i];
    lo = fminf(lo, v); hi = fmaxf(hi, v);
  }
  slo[tid] = lo; shi[tid] = hi;
  __syncthreads();
  for (int s = 512; s > 0; s >>= 1) {
    if (tid < s) {
      slo[tid] = fminf(slo[tid], slo[tid + s]);
      shi[tid] = fmaxf(shi[tid], shi[tid + s]);
    }
    __syncthreads();
  }
  lo = slo[0]; hi = shi[0];
  const float inv = 1.0f / (hi - lo + 1e-16f);
  const float w0 = lin_w[0], w1 = lin_w[1], b0 = lin_b[0], b1 = lin_b[1];
  for (int i = tid; i < nseg; i += 1024) {
    const float s = (mins[i] - lo) * inv;
    out[2 * i + 0] = s * w0 + b0;
    out[2 * i + 1] = s * w1 + b1;
  }
}

extern "C" void kernel_launch(void* const* d_in, const int* in_sizes, int n_in,
                              void* d_out, int out_size, void* d_ws, size_t ws_size,
                              hipStream_t stream) {
  const float* X   = (const float*)d_in[0];   // [1, 64, T] fp32
  const float* Q   = (const float*)d_in[1];   // [1, 64, 32] fp32
  const float* lw  = (const float*)d_in[2];   // [2, 1]
  const float* lb  = (const float*)d_in[3];   // [2]
  const int*   cum = (const int*)d_in[4];     // [nseg+1]

  const int T    = in_sizes[0] / DCH;
  const int nseg = in_sizes[4] - 1;

  float* mins = (float*)d_ws;                 // nseg floats of scratch

  shapelet_seg_min<<<nseg, BLOCK, 0, stream>>>(X, Q, cum, mins, T);
  shapelet_finalize<<<1, 1024, 0, stream>>>(mins, lw, lb, (float*)d_out, nseg);
}